// ResidualVectorQuantizer_5720896438810
// MI455X (gfx1250) — compile-verified
//
#include <hip/hip_runtime.h>
#include <hip/hip_bf16.h>
#include <math.h>

typedef float v2f __attribute__((ext_vector_type(2)));
typedef float v8f __attribute__((ext_vector_type(8)));
typedef int   v2i __attribute__((ext_vector_type(2)));

// Pointee-typed, address-space-qualified pointers for the async builtin:
// arg0 = global (AS1) int2*, arg1 = LDS (AS3) int2*  (per hipcc diagnostic).
typedef __attribute__((address_space(1))) v2i* gas_v2i_ptr;
typedef __attribute__((address_space(3))) v2i* las_v2i_ptr;

#define DIMS   128
#define KCODES 1024
#define NVEC   65536          // 16*64*64
#define CHUNK  64             // codes staged in LDS per pass
#define LSTRIDE 65            // float2 stride per dim-pair row (odd -> conflict-free stores)

// gfx1250 async global->LDS path (ASYNCcnt-tracked, no VGPR round-trip)
#if defined(__has_builtin)
#  if __has_builtin(__builtin_amdgcn_global_load_async_to_lds_b64) && \
      __has_builtin(__builtin_amdgcn_s_wait_asynccnt)
#    define USE_ASYNC_LDS 1
#  endif
#endif

// ---------------------------------------------------------------------------
// Per-level codebook norms ||E_k||^2 and histogram zeroing. Launch <<<4,256>>>.
// ---------------------------------------------------------------------------
__global__ __launch_bounds__(256)
void rvq_norms_kernel(const float* __restrict__ E,
                      float* __restrict__ norms,
                      unsigned* __restrict__ hist)
{
    int k = blockIdx.x * 256 + threadIdx.x;          // 0..1023
    const float* row = E + (size_t)k * DIMS;
    float s = 0.0f;
#pragma unroll
    for (int d = 0; d < DIMS; d += 4) {
        float4 v = *(const float4*)(row + d);
        s += v.x * v.x + v.y * v.y + v.z * v.z + v.w * v.w;
    }
    norms[k] = s;
    hist[k]  = 0u;
}

// ---------------------------------------------------------------------------
// Distance GEMM + row argmin via V_WMMA_F32_16X16X4_F32.
// Block = 256 threads = 8 waves; each wave owns a 16-row tile of the residual.
// Codebook swept in 64-code chunks staged in LDS as (dpair, code) float2.
// Two N-tiles per pass -> two independent WMMA accumulation chains; B frags
// prefetched one K-step ahead so ds latency hides behind two WMMAs.
// Launch <<<NVEC/128, 256>>>.
// ---------------------------------------------------------------------------
__global__ __launch_bounds__(256)
void rvq_argmin_kernel(const float* __restrict__ R,
                       const float* __restrict__ E,
                       const float* __restrict__ norms,
                       int* __restrict__ codes)
{
    __shared__ v2f ldsB[(DIMS / 2) * LSTRIDE];       // 64 dpairs x 65 float2 = 33,280 B

    const int tid  = threadIdx.x;
    const int lane = tid & 31;
    const int wave = tid >> 5;
    const int half = lane >> 4;                      // K-half of the wave
    const int col  = lane & 15;

    const int rowbase = blockIdx.x * 128 + wave * 16;
    const int myrow   = rowbase + col;

    // A fragments: this wave's 16 residual rows, 32 K-steps of 4 dims.
    v2f afrag[32];
    const float* rp = R + (size_t)myrow * DIMS + 2 * half;
#pragma unroll
    for (int s = 0; s < 32; ++s)
        afrag[s] = *(const v2f*)(rp + 4 * s);

    float minv[8];
    int   mini[8];
#pragma unroll
    for (int i = 0; i < 8; ++i) { minv[i] = 3.4e38f; mini[i] = 0; }

    for (int cbase = 0; cbase < KCODES; cbase += CHUNK) {
        __syncthreads();                             // LDS reusable (prev tiles done)
#if defined(USE_ASYNC_LDS)
        // Async DMA global -> LDS: one instruction per element, single wait.
#pragma unroll
        for (int it = 0; it < (CHUNK * (DIMS / 2)) / 256; ++it) {   // 16 iters
            int linear = it * 256 + tid;
            int c  = linear >> 6;                    // code within chunk
            int dp = linear & 63;                    // dim pair
            __builtin_amdgcn_global_load_async_to_lds_b64(
                (gas_v2i_ptr)(E + (size_t)(cbase + c) * DIMS + 2 * dp),
                (las_v2i_ptr)&ldsB[dp * LSTRIDE + c],
                0, 0);
        }
        __builtin_amdgcn_s_wait_asynccnt(0);
#else
        // Fallback: batch 16 loads in flight, then 16 LDS stores.
        v2f tmp[16];
#pragma unroll
        for (int it = 0; it < 16; ++it) {
            int linear = it * 256 + tid;
            int c  = linear >> 6;
            int dp = linear & 63;
            tmp[it] = *(const v2f*)(E + (size_t)(cbase + c) * DIMS + 2 * dp);
        }
#pragma unroll
        for (int it = 0; it < 16; ++it) {
            int linear = it * 256 + tid;
            int c  = linear >> 6;
            int dp = linear & 63;
            ldsB[dp * LSTRIDE + c] = tmp[it];
        }
#endif
        __syncthreads();

        for (int t = 0; t < CHUNK / 16; t += 2) {    // 2 N-tiles per pass
            const int n0 = cbase + t * 16 + col;
            const int n1 = n0 + 16;
            const float nrm0 = norms[n0];
            const float nrm1 = norms[n1];
            const v2f* bp = &ldsB[t * 16 + col];

            v8f acc0 = {}, acc1 = {};
            v2f b0 = bp[half * LSTRIDE];
            v2f b1 = bp[half * LSTRIDE + 16];
#pragma unroll
            for (int s = 0; s < 32; ++s) {
                v2f nb0 = {}, nb1 = {};
                if (s < 31) {                        // prefetch next K-step
                    nb0 = bp[(2 * (s + 1) + half) * LSTRIDE];
                    nb1 = bp[(2 * (s + 1) + half) * LSTRIDE + 16];
                }
                acc0 = __builtin_amdgcn_wmma_f32_16x16x4_f32(
                        false, afrag[s], false, b0, (short)0, acc0, false, false);
                acc1 = __builtin_amdgcn_wmma_f32_16x16x4_f32(
                        false, afrag[s], false, b1, (short)0, acc1, false, false);
                b0 = nb0; b1 = nb1;
            }
#pragma unroll
            for (int i = 0; i < 8; ++i) {
                float d0 = nrm0 - 2.0f * acc0[i];    // ||e||^2 - 2 f.e
                if (d0 < minv[i]) { minv[i] = d0; mini[i] = n0; }
                float d1 = nrm1 - 2.0f * acc1[i];
                if (d1 < minv[i]) { minv[i] = d1; mini[i] = n1; }
            }
        }
    }

    // Cross-lane argmin over the 16 columns of each half-wave (rows i / i+8).
#pragma unroll
    for (int i = 0; i < 8; ++i) {
        float v = minv[i]; int ix = mini[i];
#pragma unroll
        for (int m = 1; m <= 8; m <<= 1) {
            float ov = __shfl_xor(v, m, 32);
            int   oi = __shfl_xor(ix, m, 32);
            if (ov < v || (ov == v && oi < ix)) { v = ov; ix = oi; }
        }
        minv[i] = v; mini[i] = ix;
    }
    if (col == 0) {
#pragma unroll
        for (int i = 0; i < 8; ++i)
            codes[rowbase + half * 8 + i] = mini[i];
    }
}

// ---------------------------------------------------------------------------
// Gather q = E[codes], write q_level, residual -= q, quantized += q,
// block-reduced SSE partials (deterministic), integer histogram.
// One float4 per thread. Launch <<<NVEC*DIMS/4/256, 256>>>.
// ---------------------------------------------------------------------------
__global__ __launch_bounds__(256)
void rvq_gather_kernel(const float* __restrict__ Rin,
                       const float* __restrict__ E,
                       const int* __restrict__ codes,
                       float* __restrict__ Rout,
                       float* __restrict__ qlev,
                       float* __restrict__ quant,
                       float* __restrict__ partials,
                       unsigned* __restrict__ hist,
                       int firstLevel)
{
    __shared__ float red[256];
    const int tid = threadIdx.x;
    const int gid = blockIdx.x * 256 + tid;
    const int row = gid >> 5;                        // 32 float4 per 128-dim row
    const int dq  = (gid & 31) << 2;
    const int code = codes[row];
    const size_t off = (size_t)row * DIMS + dq;

    float4 qv = *(const float4*)(E + (size_t)code * DIMS + dq);
    float4 rv = *(const float4*)(Rin + off);

    *(float4*)(qlev + off) = qv;                     // q_st forward value == q

    float4 nr = make_float4(rv.x - qv.x, rv.y - qv.y, rv.z - qv.z, rv.w - qv.w);
    *(float4*)(Rout + off) = nr;

    if (firstLevel) {
        *(float4*)(quant + off) = qv;
    } else {
        float4 a = *(const float4*)(quant + off);
        a.x += qv.x; a.y += qv.y; a.z += qv.z; a.w += qv.w;
        *(float4*)(quant + off) = a;
    }

    float sse = nr.x * nr.x + nr.y * nr.y + nr.z * nr.z + nr.w * nr.w;
    red[tid] = sse;
    __syncthreads();
#pragma unroll
    for (int st = 128; st > 0; st >>= 1) {
        if (tid < st) red[tid] += red[tid + st];
        __syncthreads();
    }
    if (tid == 0) partials[blockIdx.x] = red[0];

    if ((gid & 31) == 0) atomicAdd(&hist[code], 1u); // one count per vector
}

// ---------------------------------------------------------------------------
// Reduce SSE partials -> loss[level]; histogram -> perplexity[level];
// at level 3 also write total_loss = mean(losses). Launch <<<1,1024>>>.
// ---------------------------------------------------------------------------
__global__ __launch_bounds__(1024)
void rvq_finalize_kernel(const float* __restrict__ partials, int nPartials,
                         const unsigned* __restrict__ hist,
                         float* __restrict__ losses,
                         float* __restrict__ out_perp,
                         float* __restrict__ out_loss,
                         int level)
{
    __shared__ float red[1024];
    const int t = threadIdx.x;

    float s = 0.0f;
    for (int i = t; i < nPartials; i += 1024) s += partials[i];
    red[t] = s;
    __syncthreads();
#pragma unroll
    for (int st = 512; st > 0; st >>= 1) {
        if (t < st) red[t] += red[t + st];
        __syncthreads();
    }
    const float sse = red[0];
    __syncthreads();

    float p = (float)hist[t] * (1.0f / (float)NVEC);
    red[t] = p * logf(p + 1e-10f);
    __syncthreads();
#pragma unroll
    for (int st = 512; st > 0; st >>= 1) {
        if (t < st) red[t] += red[t + st];
        __syncthreads();
    }

    if (t == 0) {
        losses[level]   = 0.25f * sse / (float)((size_t)NVEC * DIMS);
        out_perp[level] = expf(-red[0]);
        if (level == 3) {
            out_loss[0] = 0.25f * (losses[0] + losses[1] + losses[2] + losses[3]);
        }
    }
}

// ---------------------------------------------------------------------------
extern "C" void kernel_launch(void* const* d_in, const int* in_sizes, int n_in,
                              void* d_out, int out_size, void* d_ws, size_t ws_size,
                              hipStream_t stream)
{
    const float* x   = (const float*)d_in[0];        // [16,64,64,128]
    const float* cbs = (const float*)d_in[1];        // [4,1024,128]
    float* out = (float*)d_out;

    const size_t NDD = (size_t)NVEC * DIMS;          // 8,388,608

    float* quant    = out;                           // [NDD]
    float* out_loss = out + NDD;                     // [1]
    float* qlev     = out + NDD + 1;                 // [4, NDD]
    float* out_perp = out + NDD + 1 + 4 * NDD;       // [4]

    char* ws = (char*)d_ws;
    float*    residual = (float*)ws;                                  // NDD f32
    int*      codes    = (int*)(ws + NDD * 4);                        // NVEC i32
    float*    norms    = (float*)(ws + NDD * 4 + (size_t)NVEC * 4);   // 1024 f32
    unsigned* hist     = (unsigned*)((char*)norms + 1024 * 4);        // 1024 u32
    float*    partials = (float*)((char*)hist + 1024 * 4);            // 8192 f32
    float*    losses   = (float*)((char*)partials + 8192 * 4);        // 4 f32

    const int nGatherBlocks = (int)(NDD / 4 / 256);  // 8192

    for (int q = 0; q < 4; ++q) {
        const float* E   = cbs + (size_t)q * KCODES * DIMS;
        const float* Rin = (q == 0) ? x : residual;

        rvq_norms_kernel<<<4, 256, 0, stream>>>(E, norms, hist);
        rvq_argmin_kernel<<<NVEC / 128, 256, 0, stream>>>(Rin, E, norms, codes);
        rvq_gather_kernel<<<nGatherBlocks, 256, 0, stream>>>(
            Rin, E, codes, residual, qlev + (size_t)q * NDD, quant,
            partials, hist, q == 0);
        rvq_finalize_kernel<<<1, 1024, 0, stream>>>(
            partials, nGatherBlocks, hist, losses, out_perp, out_loss, q);
    }
}